// LabelAggregator_46411416600869
// MI455X (gfx1250) — compile-verified
//
#include <hip/hip_runtime.h>
#include <hip/hip_bf16.h>
#include <math.h>

typedef __attribute__((ext_vector_type(2))) float v2f;
typedef __attribute__((ext_vector_type(8))) float v8f;

#define B_ 16
#define L_ 2048
#define H_ 768
#define D_ 768
#define M_ 8

// ---------------------------------------------------------------------------
// gfx1250 async global->LDS copy (ASYNCcnt-tracked, CDNA5 ISA).
// One instruction moves 16B per active lane directly into LDS.
// ---------------------------------------------------------------------------
__device__ __forceinline__ unsigned int lds_off(const void* p) {
  // LDS aperture truncation: low 32 bits of a generic shared pointer are the
  // LDS byte offset (ISA 10.2 per-aperture address calculation).
  return (unsigned int)(unsigned long long)p;
}

__device__ __forceinline__ void async_copy_b128(unsigned int ldsaddr,
                                                const float* gptr) {
  asm volatile("global_load_async_to_lds_b128 %0, %1, off"
               :: "v"(ldsaddr), "v"((unsigned long long)gptr)
               : "memory");
}

__device__ __forceinline__ void wait_async0() {
  asm volatile("s_wait_asynccnt 0x0" ::: "memory");
}

// ---------------------------------------------------------------------------
// Kernel 1: segment-mean of hidden_states over label tokens -> hbar (B*M, H).
// (mean commutes with the affine label projector, so we project the mean.)
// ---------------------------------------------------------------------------
__global__ void seg_mean_kernel(const float* __restrict__ hs,
                                const int*   __restrict__ lmask,
                                float*       __restrict__ hbar) {
  int bm = blockIdx.x;               // 0..B*M-1
  int b = bm / M_, m = bm % M_;
  int tid = threadIdx.x;             // 256 threads, H = 3*256
  float a0 = 0.f, a1 = 0.f, a2 = 0.f;
  int cnt = 0;
  const int* lm = lmask + (size_t)b * L_;
  for (int l = 0; l < L_; ++l) {
    if (lm[l] == m + 1) {
      const float* h = hs + ((size_t)b * L_ + l) * H_;
      a0 += h[tid];
      a1 += h[tid + 256];
      a2 += h[tid + 512];
      ++cnt;
    }
  }
  float inv = 1.0f / fmaxf((float)cnt, 1.0f);
  float* o = hbar + (size_t)bm * H_;
  o[tid]       = a0 * inv;
  o[tid + 256] = a1 * inv;
  o[tid + 512] = a2 * inv;
}

// ---------------------------------------------------------------------------
// fp32 WMMA GEMM, LDS-staged + double-buffered via async global->LDS copies.
//   C[b] = A[b](Rvalid x K, zero-padded to 16 rows) @ W[b](K x N or N x K) + bias
// Block: 128 threads = 4 waves; 16 x 128 output tile. Each wave owns TWO
// 16x16 WMMA tiles (independent accumulator chains; one ds-loaded A fragment
// feeds both WMMAs). BLK_K = 32 -> 16 x V_WMMA_F32_16X16X4_F32 per wave/stage.
// LDS layouts chosen so every global->LDS transfer is a contiguous b128:
//   As[r][k]           (stride 36: 16 distinct banks + 16B aligned)
//   !TRANSW: Bs[k][n]  (k-major stride 128, matches W row-major)
//    TRANSW: Bs[n][k]  (n-major stride 36,  matches W row-major)
// Rows >= Rvalid are zero-filled ONCE (async copies never touch them).
// ---------------------------------------------------------------------------
template <bool TRANSW>
__global__ void __launch_bounds__(128)
gemm_wmma_f32(const float* __restrict__ A, long aBS, int lda, int Rvalid,
              const float* __restrict__ W, long wBS, int wld,
              const float* __restrict__ bias,
              float* __restrict__ C, long cBS, int ldc, int K) {
  constexpr int SA = 36;                    // padded row stride (floats)
  constexpr int SB = 36;
  __shared__ __align__(16) float As[2][16 * SA];
  __shared__ __align__(16) float Bs[2][128 * SB];

  const int tid   = threadIdx.x;
  const int wave  = tid >> 5;
  const int lane  = tid & 31;
  const int idx   = lane & 15;
  const int half  = lane >> 4;
  const int kbase = half * 2;

  const int row0 = blockIdx.x * 16;
  const int col0 = blockIdx.y * 128;
  const int bz   = blockIdx.z;
  const float* Ab = A + (size_t)bz * aBS;
  const float* Wb = W + (size_t)bz * wBS;
  float*       Cb = C + (size_t)bz * cBS;

  const int ar   = tid >> 3;                // A stage: row 0..15
  const int ac4  = (tid & 7) * 4;           // A stage: col group
  const bool arow_ok = (row0 + ar < Rvalid);

  // One-time zero fill of pad rows in both buffers (loop never rewrites them).
  if (!arow_ok) {
    float4 z = make_float4(0.f, 0.f, 0.f, 0.f);
    *(float4*)&As[0][ar * SA + ac4] = z;
    *(float4*)&As[1][ar * SA + ac4] = z;
  }

  auto stage = [&](int buf, int kk) {
    // ---- A tile: 16 x 32, one b128 per valid-row thread ----
    if (arow_ok) {
      async_copy_b128(lds_off(&As[buf][ar * SA + ac4]),
                      Ab + (size_t)(row0 + ar) * lda + kk + ac4);
    }
    // ---- B tile: 32 x 128 (or 128 x 32), 8 b128 per thread ----
#pragma unroll
    for (int j = 0; j < 8; ++j) {
      const int id = tid + 128 * j;
      if (TRANSW) {
        const int n = id >> 3, c4 = (id & 7) * 4;
        async_copy_b128(lds_off(&Bs[buf][n * SB + c4]),
                        Wb + (size_t)(col0 + n) * wld + kk + c4);
      } else {
        const int k = id >> 5, c4 = (id & 31) * 4;
        async_copy_b128(lds_off(&Bs[buf][k * 128 + c4]),
                        Wb + (size_t)(kk + k) * wld + col0 + c4);
      }
    }
  };

  v8f acc0 = {};
  v8f acc1 = {};
  stage(0, 0);
  int buf = 0;
  for (int kk = 0; kk < K; kk += 32) {
    wait_async0();                      // staged copies for `buf` complete
    __syncthreads();                    // visible to all waves
    if (kk + 32 < K) stage(buf ^ 1, kk + 32);   // overlap next stage

    const float* a_base  = &As[buf][idx * SA];
    const float* b_base0 = TRANSW ? &Bs[buf][(wave * 16 + idx) * SB]
                                  : &Bs[buf][wave * 16 + idx];
    const float* b_base1 = TRANSW ? b_base0 + 64 * SB
                                  : b_base0 + 64;
#pragma unroll
    for (int k4 = 0; k4 < 32; k4 += 4) {
      const int k0 = k4 + kbase;
      v2f a = *(const v2f*)(a_base + k0);           // ds_load_b64 (shared)
      v2f b0, b1;
      if (TRANSW) {
        b0 = *(const v2f*)(b_base0 + k0);           // ds_load_b64
        b1 = *(const v2f*)(b_base1 + k0);
      } else {
        b0.x = b_base0[(size_t)k0 * 128];           // ds_load_b32 pairs
        b0.y = b_base0[(size_t)(k0 + 1) * 128];
        b1.x = b_base1[(size_t)k0 * 128];
        b1.y = b_base1[(size_t)(k0 + 1) * 128];
      }
      // two independent accumulator chains per wave
      acc0 = __builtin_amdgcn_wmma_f32_16x16x4_f32(false, a, false, b0,
                                                   (short)0, acc0, false, false);
      acc1 = __builtin_amdgcn_wmma_f32_16x16x4_f32(false, a, false, b1,
                                                   (short)0, acc1, false, false);
    }
    __syncthreads();                    // all reads of `buf` done
    buf ^= 1;
  }

  const int mbase = row0 + half * 8;
#pragma unroll
  for (int t = 0; t < 2; ++t) {
    const int   col = col0 + wave * 16 + t * 64 + idx;
    const float bv  = bias ? bias[col] : 0.0f;
    const v8f&  acc = t ? acc1 : acc0;
#pragma unroll
    for (int r = 0; r < 8; ++r) {
      const int row = mbase + r;
      if (row < Rvalid)
        Cb[(size_t)row * ldc + col] = acc[r] + bv;
    }
  }
}

// ---------------------------------------------------------------------------
// Kernel: cb[b,m] = dot(agg[b,m], bt)   (the bias term folded out of scores)
// ---------------------------------------------------------------------------
__global__ void cbias_kernel(const float* __restrict__ agg,
                             const float* __restrict__ bt,
                             float* __restrict__ cb) {
  int w    = (int)((blockIdx.x * blockDim.x + threadIdx.x) >> 5);
  int lane = threadIdx.x & 31;
  if (w >= B_ * M_) return;
  const float* a = agg + (size_t)w * D_;
  float s = 0.f;
  for (int k = lane; k < D_; k += 32) s += a[k] * bt[k];
  for (int off = 16; off; off >>= 1) s += __shfl_xor(s, off, 32);
  if (lane == 0) cb[w] = s;
}

// ---------------------------------------------------------------------------
// Kernel: scores[b,m,l] = (q[b,m].hs[b,l] + cb[b,m]) / temp, -inf off-text.
// One wave per score; float4 lane-strided coalesced dot (global_load_b128),
// shuffle-reduced. Streaming pass over hs (L2-resident: 101MB << 192MB L2).
// ---------------------------------------------------------------------------
__global__ void scores_kernel(const float* __restrict__ hs,
                              const float* __restrict__ q,
                              const float* __restrict__ cb,
                              const int*   __restrict__ lmask,
                              const int*   __restrict__ amask,
                              const float* __restrict__ temp_p,
                              float* __restrict__ S) {
  long gw  = (long)((blockIdx.x * (long)blockDim.x + threadIdx.x) >> 5);
  int lane = threadIdx.x & 31;
  if (gw >= (long)B_ * M_ * L_) return;
  int l  = (int)(gw % L_);
  int bm = (int)(gw / L_);
  int b  = bm / M_;
  const float4* h4 = (const float4*)(hs + ((size_t)b * L_ + l) * H_);
  const float4* q4 = (const float4*)(q + (size_t)bm * H_);
  float s = 0.f;
#pragma unroll
  for (int k = lane; k < H_ / 4; k += 32) {       // 6 iterations
    float4 a = h4[k], w = q4[k];
    s += a.x * w.x + a.y * w.y + a.z * w.z + a.w * w.w;
  }
  for (int off = 16; off; off >>= 1) s += __shfl_xor(s, off, 32);
  if (lane == 0) {
    float t = fmaxf(fabsf(temp_p[0]), 0.1f);
    int li = b * L_ + l;
    bool text = (lmask[li] == 0) && (amask[li] == 1);
    S[(size_t)bm * L_ + l] = text ? (s + cb[bm]) / t : -__builtin_inff();
  }
}

// ---------------------------------------------------------------------------
// Kernel: row softmax over L in place (one 256-thread block per (b,m) row).
// ---------------------------------------------------------------------------
__global__ void softmax_kernel(float* __restrict__ S) {
  int bm = blockIdx.x;
  float* row = S + (size_t)bm * L_;
  int tid = threadIdx.x;              // 256
  __shared__ float red[256];

  float mx = -__builtin_inff();
  for (int l = tid; l < L_; l += 256) mx = fmaxf(mx, row[l]);
  red[tid] = mx;
  __syncthreads();
  for (int s = 128; s; s >>= 1) {
    if (tid < s) red[tid] = fmaxf(red[tid], red[tid + s]);
    __syncthreads();
  }
  mx = red[0];
  __syncthreads();

  bool finite = (mx > -__builtin_inff());
  float sum = 0.f;
  for (int l = tid; l < L_; l += 256) {
    float e = finite ? expf(row[l] - mx) : 0.0f;
    row[l] = e;
    sum += e;
  }
  red[tid] = sum;
  __syncthreads();
  for (int s = 128; s; s >>= 1) {
    if (tid < s) red[tid] += red[tid + s];
    __syncthreads();
  }
  float inv = 1.0f / fmaxf(red[0], 1e-30f);
  __syncthreads();
  for (int l = tid; l < L_; l += 256) row[l] *= inv;
}

// ---------------------------------------------------------------------------
// Kernel: cosine head -> logits, ids, scale. One wave per (b,m) row pair.
// ---------------------------------------------------------------------------
__global__ void finalize_kernel(const float* __restrict__ at,
                                const float* __restrict__ al,
                                const float* __restrict__ logit_scale,
                                float* __restrict__ out) {
  int w    = (int)((blockIdx.x * blockDim.x + threadIdx.x) >> 5);
  int lane = threadIdx.x & 31;
  if (w >= B_ * M_) return;
  const float* x = at + (size_t)w * D_;
  const float* y = al + (size_t)w * D_;
  float dd = 0.f, nx = 0.f, ny = 0.f;
  for (int k = lane; k < D_; k += 32) {
    float xv = x[k], yv = y[k];
    dd += xv * yv;
    nx += xv * xv;
    ny += yv * yv;
  }
  for (int off = 16; off; off >>= 1) {
    dd += __shfl_xor(dd, off, 32);
    nx += __shfl_xor(nx, off, 32);
    ny += __shfl_xor(ny, off, 32);
  }
  if (lane == 0) {
    float scale = expf(logit_scale[0]);
    float denom = fmaxf(sqrtf(nx), 1e-8f) * fmaxf(sqrtf(ny), 1e-8f);
    out[w]       = (dd / denom) * scale;          // logits (B*M,1)
    out[128 + w] = (float)(w / M_);               // all_batch_ids
    out[256 + w] = (float)(w % M_ + 1);           // all_label_ids
    if (w == 0) out[98688] = scale;               // scale scalar
  }
}

// ---------------------------------------------------------------------------
extern "C" void kernel_launch(void* const* d_in, const int* in_sizes, int n_in,
                              void* d_out, int out_size, void* d_ws, size_t ws_size,
                              hipStream_t stream) {
  const float* hs    = (const float*)d_in[0];   // (B,L,H)
  const float* Wt    = (const float*)d_in[1];   // (H,D)
  const float* bt    = (const float*)d_in[2];   // (D)
  const float* Wl    = (const float*)d_in[3];   // (H,D)
  const float* bl    = (const float*)d_in[4];   // (D)
  const float* temp  = (const float*)d_in[5];   // scalar
  const float* lsc   = (const float*)d_in[6];   // scalar
  const int*   lmask = (const int*)d_in[7];     // (B,L)
  const int*   amask = (const int*)d_in[9];     // (B,L)
  (void)in_sizes; (void)n_in; (void)out_size; (void)ws_size;

  float* ws   = (float*)d_ws;
  float* hbar = ws;               // 128*768
  float* q    = ws + 98304;       // 128*768
  float* cb   = ws + 196608;      // 128
  float* t    = ws + 196736;      // 128*768
  float* S    = ws + 295040;      // 16*8*2048

  float* out = (float*)d_out;
  float* agg = out + 384;         // aggregated_labels (B*M, D)
  float* atx = out + 98689;       // aggregated_text   (B*M, D)

  // 1) segment-mean of hidden over label tokens
  seg_mean_kernel<<<B_ * M_, 256, 0, stream>>>(hs, lmask, hbar);

  // 2) aggregated_labels = hbar @ Wl + bl            (128 x 768 x 768, WMMA)
  gemm_wmma_f32<false><<<dim3(8, 6, 1), 128, 0, stream>>>(
      hbar, 0, H_, 128, Wl, 0, D_, bl, agg, 0, D_, H_);

  // 3) q = agg @ Wt^T                                 (128 x 768 x 768, WMMA)
  gemm_wmma_f32<true><<<dim3(8, 6, 1), 128, 0, stream>>>(
      agg, 0, D_, 128, Wt, 0, D_, nullptr, q, 0, H_, D_);

  // 4) cb = agg . bt  (score bias term)
  cbias_kernel<<<32, 128, 0, stream>>>(agg, bt, cb);

  // 5) scores over raw hs (proj_text eliminated algebraically)
  scores_kernel<<<(B_ * M_ * L_) / 8, 256, 0, stream>>>(
      hs, q, cb, lmask, amask, temp, S);

  // 6) softmax over text tokens
  softmax_kernel<<<B_ * M_, 256, 0, stream>>>(S);

  // 7) t[b] = attn[b] @ hs[b]           (batched 8 x 2048 x 768, WMMA, padded)
  gemm_wmma_f32<false><<<dim3(1, 6, B_), 128, 0, stream>>>(
      S, (long)M_ * L_, L_, 8, hs, (long)L_ * H_, H_, nullptr,
      t, (long)M_ * H_, H_, L_);

  // 8) aggregated_text = t @ Wt + bt                  (128 x 768 x 768, WMMA)
  gemm_wmma_f32<false><<<dim3(8, 6, 1), 128, 0, stream>>>(
      t, 0, H_, 128, Wt, 0, D_, bt, atx, 0, D_, H_);

  // 9) cosine head + ids + scale
  finalize_kernel<<<32, 128, 0, stream>>>(atx, agg, lsc, out);
}